// GraphEncoder_22273700397650
// MI455X (gfx1250) — compile-verified
//
#include <hip/hip_runtime.h>

#define NN 65536
#define EE 1048576
#define BB 64
#define BN_EPS 1e-5f

typedef __attribute__((ext_vector_type(2))) float v2f;
typedef __attribute__((ext_vector_type(8))) float v8f;

// ---------------- small utility kernels ----------------
__global__ __launch_bounds__(256) void k_fill1(float* __restrict__ p) {
    p[blockIdx.x * 256 + threadIdx.x] = 1.0f;          // self-loop contributes 1 to degree
}

__global__ __launch_bounds__(256) void k_zero4(float4* __restrict__ p) {
    p[blockIdx.x * 256 + threadIdx.x] = make_float4(0.f, 0.f, 0.f, 0.f);
}

__global__ __launch_bounds__(256) void k_zero_n(float* __restrict__ p, int n) {
    int i = blockIdx.x * 256 + threadIdx.x;
    if (i < n) p[i] = 0.f;
}

__global__ __launch_bounds__(256) void k_deg_edges(const int* __restrict__ col,
                                                   float* __restrict__ deg) {
    int e = blockIdx.x * 256 + threadIdx.x;
    atomicAdd(&deg[col[e]], 1.0f);
}

__global__ __launch_bounds__(256) void k_dis(const float* __restrict__ deg,
                                             float* __restrict__ dis) {
    int i = blockIdx.x * 256 + threadIdx.x;
    dis[i] = rsqrtf(deg[i]);                           // deg >= 1 always (self-loops)
}

__global__ __launch_bounds__(256) void k_norm(const int* __restrict__ row,
                                              const int* __restrict__ col,
                                              const float* __restrict__ dis,
                                              float* __restrict__ nrm) {
    int e = blockIdx.x * 256 + threadIdx.x;
    nrm[e] = dis[row[e]] * dis[col[e]];
}

// ---------------- H = X @ W via V_WMMA_F32_16X16X4_F32 ----------------
// X: [NN,64] row-major, W: [64,64] row-major, H: [NN,64].
// One wave -> 16 rows x 64 cols (4 accumulators), 8 waves/block -> 128 rows/block.
// W is staged in LDS pre-swizzled into the exact WMMA B-fragment lane layout:
//   Ws[kstep*256 + lane*8 + nt*2 + j] = W[(kstep*4 + 2*(lane>>4) + j)*64 + nt*16 + (lane&15)]
// so each lane fetches its 8 B floats per k-step with two aligned ds_load_b128,
// with the v2f pairs already register-contiguous (no shuffle movs).
__global__ __launch_bounds__(256) void k_gemm64(const float* __restrict__ X,
                                                const float* __restrict__ W,
                                                float* __restrict__ H) {
    __shared__ float Ws[16 * 256];
#pragma unroll
    for (int it = 0; it < 16; ++it) {
        int idx   = it * 256 + threadIdx.x;
        int kstep = idx >> 8;
        int r     = idx & 255;
        int ln    = r >> 3;
        int q     = r & 7;
        int nt    = q >> 1;
        int j     = q & 1;
        int k     = kstep * 4 + ((ln >> 4) << 1) + j;
        int c     = nt * 16 + (ln & 15);
        Ws[idx] = W[k * 64 + c];
    }
    __syncthreads();

    const int lane    = threadIdx.x & 31;
    const int wave    = threadIdx.x >> 5;
    const int rowBase = (blockIdx.x * 8 + wave) * 16;
    const int n       = lane & 15;                // N index within 16-col tile / M for A frag
    const int kh      = (lane >> 4) << 1;         // 0 or 2 (K split across lane halves)
    const unsigned m  = (unsigned)(rowBase + n);  // A fragment row
    const float* wl   = Ws + lane * 8;            // this lane's fragment base

    v8f acc0 = {}, acc1 = {}, acc2 = {}, acc3 = {};
#pragma unroll
    for (int kstep = 0; kstep < 16; ++kstep) {
        const int k = kstep * 4 + kh;
        v2f a  = *(const v2f*)(X + m * 64u + (unsigned)k);     // 8B-aligned (k even)
        v2f b0 = *(const v2f*)(wl + kstep * 256 + 0);          // 16B-aligned pairs
        v2f b1 = *(const v2f*)(wl + kstep * 256 + 2);
        v2f b2 = *(const v2f*)(wl + kstep * 256 + 4);
        v2f b3 = *(const v2f*)(wl + kstep * 256 + 6);
        acc0 = __builtin_amdgcn_wmma_f32_16x16x4_f32(false, a, false, b0, (short)0, acc0, false, false);
        acc1 = __builtin_amdgcn_wmma_f32_16x16x4_f32(false, a, false, b1, (short)0, acc1, false, false);
        acc2 = __builtin_amdgcn_wmma_f32_16x16x4_f32(false, a, false, b2, (short)0, acc2, false, false);
        acc3 = __builtin_amdgcn_wmma_f32_16x16x4_f32(false, a, false, b3, (short)0, acc3, false, false);
    }

    const int rh = (lane >> 4) << 3;              // C/D layout: lanes 16-31 hold M+8
#pragma unroll
    for (int r = 0; r < 8; ++r) {
        unsigned mm = (unsigned)(rowBase + rh + r);
        H[mm * 64u + n]      = acc0[r];
        H[mm * 64u + n + 16] = acc1[r];
        H[mm * 64u + n + 32] = acc2[r];
        H[mm * 64u + n + 48] = acc3[r];
    }
}

// ---------------- edge scatter: agg[col] += h[row] * norm ----------------
// 16 lanes per edge, float4 chunks, f32 global atomics (L2-resident working set).
__global__ __launch_bounds__(256) void k_scatter(const float* __restrict__ h,
                                                 const int* __restrict__ row,
                                                 const int* __restrict__ col,
                                                 const float* __restrict__ nrm,
                                                 float* __restrict__ agg) {
    unsigned t = blockIdx.x * 256u + threadIdx.x;      // EE*16 threads
    unsigned e = t >> 4;
    unsigned c = (t & 15u) << 2;
    int r = row[e];
    int d = col[e];
    float nv = nrm[e];
    float4 v = *(const float4*)(h + (unsigned)r * 64u + c);
    float* dst = agg + (unsigned)d * 64u + c;
    atomicAdd(dst + 0, v.x * nv);
    atomicAdd(dst + 1, v.y * nv);
    atomicAdd(dst + 2, v.z * nv);
    atomicAdd(dst + 3, v.w * nv);
}

// ---------------- finalize: self-loop + bias + BN(eval) + ReLU (float4) ----------------
__global__ __launch_bounds__(256) void k_finalize(const float* __restrict__ agg,
                                                  const float* __restrict__ h,
                                                  const float* __restrict__ dis,
                                                  const float* __restrict__ b,
                                                  const float* __restrict__ g,
                                                  const float* __restrict__ be,
                                                  const float* __restrict__ rm,
                                                  const float* __restrict__ rv,
                                                  float* __restrict__ y) {
    unsigned t = blockIdx.x * 256u + threadIdx.x;      // NN*16 threads
    unsigned i = t >> 4;
    unsigned c = (t & 15u) << 2;
    float ds = dis[i];
    float d2 = ds * ds;
    float4 va = *(const float4*)(agg + i * 64u + c);
    float4 vh = *(const float4*)(h   + i * 64u + c);
    float4 r;
    {
        float v0 = va.x + vh.x * d2 + b[c + 0];
        float v1 = va.y + vh.y * d2 + b[c + 1];
        float v2 = va.z + vh.z * d2 + b[c + 2];
        float v3 = va.w + vh.w * d2 + b[c + 3];
        r.x = fmaxf((v0 - rm[c + 0]) * rsqrtf(rv[c + 0] + BN_EPS) * g[c + 0] + be[c + 0], 0.f);
        r.y = fmaxf((v1 - rm[c + 1]) * rsqrtf(rv[c + 1] + BN_EPS) * g[c + 1] + be[c + 1], 0.f);
        r.z = fmaxf((v2 - rm[c + 2]) * rsqrtf(rv[c + 2] + BN_EPS) * g[c + 2] + be[c + 2], 0.f);
        r.w = fmaxf((v3 - rm[c + 3]) * rsqrtf(rv[c + 3] + BN_EPS) * g[c + 3] + be[c + 3], 0.f);
    }
    *(float4*)(y + i * 64u + c) = r;
}

// ---------------- mean pool (sums + counts via atomics) ----------------
__global__ __launch_bounds__(256) void k_pool(const float* __restrict__ y,
                                              const int* __restrict__ batch,
                                              float* __restrict__ pooled,
                                              float* __restrict__ cnt) {
    unsigned t = blockIdx.x * 256u + threadIdx.x;      // NN*16 threads
    unsigned i = t >> 4;
    unsigned c = (t & 15u) << 2;
    int bg = batch[i];
    float4 v = *(const float4*)(y + i * 64u + c);
    float* dst = pooled + (unsigned)bg * 64u + c;
    atomicAdd(dst + 0, v.x);
    atomicAdd(dst + 1, v.y);
    atomicAdd(dst + 2, v.z);
    atomicAdd(dst + 3, v.w);
    if (c == 0) atomicAdd(&cnt[bg], 1.0f);
}

// ---------------- final projection: out = (pooled/cnt) @ Wp + bp ----------------
__global__ __launch_bounds__(256) void k_proj(const float* __restrict__ pooled,
                                              const float* __restrict__ cnt,
                                              const float* __restrict__ Wp,
                                              const float* __restrict__ bp,
                                              float* __restrict__ out) {
    int t = blockIdx.x * 256 + threadIdx.x;            // BB*64 threads
    int bg = t >> 6;
    int o  = t & 63;
    float inv = 1.0f / fmaxf(cnt[bg], 1.0f);
    float s = 0.0f;
#pragma unroll 8
    for (int j = 0; j < 64; ++j) s += pooled[bg * 64 + j] * Wp[j * 64 + o];
    out[t] = s * inv + bp[o];
}

extern "C" void kernel_launch(void* const* d_in, const int* in_sizes, int n_in,
                              void* d_out, int out_size, void* d_ws, size_t ws_size,
                              hipStream_t stream) {
    const float* x     = (const float*)d_in[0];
    const int*   ei    = (const int*)d_in[1];
    const int*   row   = ei;            // edge_index[0]
    const int*   col   = ei + EE;       // edge_index[1]
    const int*   batch = (const int*)d_in[2];
    const float* W[3]  = {(const float*)d_in[3],  (const float*)d_in[9],  (const float*)d_in[15]};
    const float* bi[3] = {(const float*)d_in[4],  (const float*)d_in[10], (const float*)d_in[16]};
    const float* g[3]  = {(const float*)d_in[5],  (const float*)d_in[11], (const float*)d_in[17]};
    const float* be[3] = {(const float*)d_in[6],  (const float*)d_in[12], (const float*)d_in[18]};
    const float* rm[3] = {(const float*)d_in[7],  (const float*)d_in[13], (const float*)d_in[19]};
    const float* rv[3] = {(const float*)d_in[8],  (const float*)d_in[14], (const float*)d_in[20]};
    const float* Wp    = (const float*)d_in[21];
    const float* bp    = (const float*)d_in[22];
    float* out = (float*)d_out;

    float* ws  = (float*)d_ws;
    float* h      = ws;  ws += (size_t)NN * 64;   // 16 MB
    float* agg    = ws;  ws += (size_t)NN * 64;   // 16 MB
    float* y      = ws;  ws += (size_t)NN * 64;   // 16 MB (reused as every layer's output)
    float* deg    = ws;  ws += NN;
    float* dis    = ws;  ws += NN;
    float* nrm    = ws;  ws += EE;                // 4 MB
    float* pooled = ws;  ws += BB * 64;
    float* cnt    = ws;  ws += BB;

    // Degrees (self-loop baked in as 1.0), dis = rsqrt(deg), per-edge norm.
    k_fill1    <<<NN / 256, 256, 0, stream>>>(deg);
    k_deg_edges<<<EE / 256, 256, 0, stream>>>(col, deg);
    k_dis      <<<NN / 256, 256, 0, stream>>>(deg, dis);
    k_norm     <<<EE / 256, 256, 0, stream>>>(row, col, dis, nrm);

    const float* in = x;
    for (int l = 0; l < 3; ++l) {
        k_gemm64  <<<NN / 128,          256, 0, stream>>>(in, W[l], h);
        k_zero4   <<<(NN * 16) / 256,   256, 0, stream>>>((float4*)agg);
        k_scatter <<<(EE * 16) / 256,   256, 0, stream>>>(h, row, col, nrm, agg);
        k_finalize<<<(NN * 16) / 256,   256, 0, stream>>>(agg, h, dis, bi[l], g[l], be[l], rm[l], rv[l], y);
        in = y;
    }

    k_zero_n<<<(BB * 64 + BB + 255) / 256, 256, 0, stream>>>(pooled, BB * 64 + BB); // pooled + cnt contiguous
    k_pool  <<<(NN * 16) / 256, 256, 0, stream>>>(y, batch, pooled, cnt);
    k_proj  <<<(BB * 64) / 256, 256, 0, stream>>>(pooled, cnt, Wp, bp, out);
}